// LSTMModel_17042430230859
// MI455X (gfx1250) — compile-verified
//
#include <hip/hip_runtime.h>

typedef __attribute__((ext_vector_type(2))) float v2f;
typedef __attribute__((ext_vector_type(8))) float v8f;

#define WPB  4      // waves per block
#define HSTR 9      // padded LDS row stride (16 rows x 8 h-values)

__device__ __forceinline__ float sigm(float x) {
    // 1 / (1 + 2^(-x*log2e)) : one v_exp_f32 + one v_rcp_f32
    return __builtin_amdgcn_rcpf(1.0f + __builtin_amdgcn_exp2f(-1.44269504f * x));
}
__device__ __forceinline__ float tanh_f(float x) {
    // tanh(x) = 2*sigmoid(2x) - 1
    return 2.0f * __builtin_amdgcn_rcpf(1.0f + __builtin_amdgcn_exp2f(-2.88539008f * x)) - 1.0f;
}
// lane i <- lane (i+8) mod 16 within each 16-lane row == xor-8 exchange,
// as a pure-VALU v_mov_b32_dpp (row_ror:8) instead of ds_bpermute.
__device__ __forceinline__ float xor8_dpp(float v) {
    int r = __builtin_amdgcn_update_dpp(0, __builtin_bit_cast(int, v),
                                        0x128 /*row_ror:8*/, 0xF, 0xF, true);
    return __builtin_bit_cast(float, r);
}

__global__ void __launch_bounds__(32 * WPB)
lstm_wmma_kernel(const float* __restrict__ x,
                 const float* __restrict__ W_ih, const float* __restrict__ W_hh,
                 const float* __restrict__ b_ih, const float* __restrict__ b_hh,
                 const float* __restrict__ W_fc, const float* __restrict__ b_fc,
                 float* __restrict__ out, int T)
{
    __shared__ float hbuf[WPB][16 * HSTR];

    const int lane = threadIdx.x & 31;
    const int wv   = threadIdx.x >> 5;
    const int n    = lane & 15;          // N column (C/D) or M row (A)
    const int half = lane >> 4;          // lane half selects K/M offset
    const int k2   = 2 * half;
    const int b0   = (blockIdx.x * WPB + wv) * 16;   // batch tile base

    // ---- B operands (weights) in 16x16x4 f32 B-layout: VGPR v, lane l -> [K=v+2*half][N=n(+16)]
    v2f BxT0, BxT1, Bh0T0, Bh1T0, Bh0T1, Bh1T1;
    BxT0.x  = W_ih[n * 4 + k2];          BxT0.y  = W_ih[n * 4 + k2 + 1];
    BxT1.x  = W_ih[(n + 16) * 4 + k2];   BxT1.y  = W_ih[(n + 16) * 4 + k2 + 1];
    Bh0T0.x = W_hh[n * 8 + k2];          Bh0T0.y = W_hh[n * 8 + k2 + 1];
    Bh1T0.x = W_hh[n * 8 + k2 + 4];      Bh1T0.y = W_hh[n * 8 + k2 + 5];
    Bh0T1.x = W_hh[(n + 16) * 8 + k2];   Bh0T1.y = W_hh[(n + 16) * 8 + k2 + 1];
    Bh1T1.x = W_hh[(n + 16) * 8 + k2+4]; Bh1T1.y = W_hh[(n + 16) * 8 + k2 + 5];
    const float bias0 = b_ih[n]      + b_hh[n];        // per-N bias, tile0
    const float bias1 = b_ih[n + 16] + b_hh[n + 16];   // tile1

    // lanes n<8 need tanh(acc1) (g), lanes n>=8 need sigmoid(acc1) (o);
    // tanh(x) = 2*sigm(2x)-1 lets one exp+rcp serve both via a per-lane pre-scale.
    const float gsel = (n < 8) ? 2.0f : 1.0f;

    // x row pointer: one b64 load per step lands directly in A-layout (K = 2*half + {0,1})
    const float* xrow = x + (size_t)(b0 + n) * (size_t)T * 4 + k2;
    float* hb = &hbuf[wv][0];

    // state: c in D-layout (valid on lanes n<8); h as two A-operands (K 0..3 / 4..7)
    v8f cst;
    #pragma unroll
    for (int i = 0; i < 8; ++i) cst[i] = 0.0f;
    v2f Ah03; Ah03.x = 0.0f; Ah03.y = 0.0f;
    v2f Ah47 = Ah03;

    // software-pipelined x load: issue t+1's load at the top of iteration t
    v2f Ax = *(const v2f*)(xrow);
    for (int t = 0; t < T; ++t) {
        const v2f Axc = Ax;
        const int tn = (t + 1 < T) ? (t + 1) : t;          // clamped, branchless prefetch
        Ax = *(const v2f*)(xrow + (size_t)tn * 4);

        v8f acc0, acc1;
        #pragma unroll
        for (int i = 0; i < 8; ++i) { acc0[i] = bias0; acc1[i] = bias1; }

        // gates[16B x 32G] = bias + x_t @ W_ih^T + h @ W_hh^T   (6x v_wmma_f32_16x16x4_f32)
        acc0 = __builtin_amdgcn_wmma_f32_16x16x4_f32(false, Axc,  false, BxT0,  (short)0, acc0, false, false);
        acc1 = __builtin_amdgcn_wmma_f32_16x16x4_f32(false, Axc,  false, BxT1,  (short)0, acc1, false, false);
        acc0 = __builtin_amdgcn_wmma_f32_16x16x4_f32(false, Ah03, false, Bh0T0, (short)0, acc0, false, false);
        acc1 = __builtin_amdgcn_wmma_f32_16x16x4_f32(false, Ah03, false, Bh0T1, (short)0, acc1, false, false);
        acc0 = __builtin_amdgcn_wmma_f32_16x16x4_f32(false, Ah47, false, Bh1T0, (short)0, acc0, false, false);
        acc1 = __builtin_amdgcn_wmma_f32_16x16x4_f32(false, Ah47, false, Bh1T1, (short)0, acc1, false, false);

        // elementwise LSTM cell update; i/g live on lanes n<8, f/o 8 lanes over
        v8f hnew;
        #pragma unroll
        for (int v = 0; v < 8; ++v) {
            float s0 = sigm(acc0[v]);              // i (n<8) / f (n>=8)
            float u  = sigm(gsel * acc1[v]);       // sigm(2*gpre) on n<8, sigm(opre) on n>=8
            float f  = xor8_dpp(s0);               // f for lanes n<8 (v_mov_b32_dpp row_ror:8)
            float o  = xor8_dpp(u);                // o for lanes n<8
            float g  = 2.0f * u - 1.0f;            // tanh(gpre) on n<8
            float cn = f * cst[v] + s0 * g;
            cst[v]   = cn;
            hnew[v]  = o * tanh_f(cn);             // valid on lanes n<8
        }

        // transpose h: D-layout -> LDS -> A-layout for next step
        if (n < 8) {
            #pragma unroll
            for (int v = 0; v < 8; ++v)
                hb[(v + 8 * half) * HSTR + n] = hnew[v];   // h[batch=v+8*half][j=n]
        }
        __builtin_amdgcn_wave_barrier();
        asm volatile("" ::: "memory");   // same-wave DS ops are in-order; just block IR reordering
        Ah03.x = hb[n * HSTR + k2];
        Ah03.y = hb[n * HSTR + k2 + 1];
        Ah47.x = hb[n * HSTR + k2 + 4];
        Ah47.y = hb[n * HSTR + k2 + 5];
        __builtin_amdgcn_wave_barrier();
    }

    // final FC: out[b] = h_last . W_fc + b_fc  (h_last already staged in LDS)
    if (half == 0) {
        float acc = b_fc[0];
        #pragma unroll
        for (int j = 0; j < 8; ++j)
            acc += hb[n * HSTR + j] * W_fc[j];
        out[b0 + n] = acc;
    }
}

extern "C" void kernel_launch(void* const* d_in, const int* in_sizes, int n_in,
                              void* d_out, int out_size, void* d_ws, size_t ws_size,
                              hipStream_t stream) {
    const float* x    = (const float*)d_in[0];
    const float* W_ih = (const float*)d_in[1];
    const float* W_hh = (const float*)d_in[2];
    const float* b_ih = (const float*)d_in[3];
    const float* b_hh = (const float*)d_in[4];
    const float* W_fc = (const float*)d_in[5];
    const float* b_fc = (const float*)d_in[6];
    float* out = (float*)d_out;

    const int T = 512;
    const int I = 4;
    const int B = in_sizes[0] / (T * I);      // 8192
    const int wavesTotal = B / 16;            // 512 waves, 16 batch rows each
    const int blocks = wavesTotal / WPB;      // 128 blocks of 4 waves

    lstm_wmma_kernel<<<blocks, 32 * WPB, 0, stream>>>(x, W_ih, W_hh, b_ih, b_hh,
                                                      W_fc, b_fc, out, T);
}